// SAGEConv_4776003633674
// MI455X (gfx1250) — compile-verified
//
#include <hip/hip_runtime.h>
#include <hip/hip_bf16.h>

#define NN 50000
#define NE 600000
#define DIM 128

typedef __attribute__((ext_vector_type(2))) float v2f;
typedef __attribute__((ext_vector_type(8))) float v8f;

// ---------------- Kernel A: zero msg_sum + deg scratch ----------------
__global__ void sage_zero_kernel(float4* __restrict__ ws, int n4) {
    int i = blockIdx.x * blockDim.x + threadIdx.x;
    if (i < n4) ws[i] = make_float4(0.f, 0.f, 0.f, 0.f);
}

// ---------------- Kernel B: edge scatter (mean-agg numerator + degree) ----
// One wave32 per edge; lane L handles floats [4L, 4L+4) of the 128-dim row.
__global__ void sage_scatter_kernel(const float* __restrict__ feat,
                                    const int* __restrict__ src,
                                    const int* __restrict__ dst,
                                    float* __restrict__ msg,
                                    float* __restrict__ deg) {
    int wave = (blockIdx.x * blockDim.x + threadIdx.x) >> 5;
    int lane = threadIdx.x & 31;
    if (wave >= NE) return;
    int s = src[wave];
    int d = dst[wave];
    const float4 v = *(const float4*)(feat + (size_t)s * DIM + lane * 4);
    float* o = msg + (size_t)d * DIM + lane * 4;
    atomicAdd(o + 0, v.x);
    atomicAdd(o + 1, v.y);
    atomicAdd(o + 2, v.z);
    atomicAdd(o + 3, v.w);
    if (lane == 0) atomicAdd(deg + d, 1.0f);
}

// ---------------- Kernel C: dual f32 WMMA GEMM + combine epilogue ---------
// Each wave computes a 16x16 tile of rst for both the self and neigh GEMMs:
//   acc_s = feat_tile    @ W_self^T   (K=128, 32 steps of V_WMMA_F32_16X16X4_F32)
//   acc_n = msg_sum_tile @ W_neigh^T
//   out   = self_w[m] * (acc_s + b_self[n]) + acc_n / max(deg[m],1) + b_neigh[n]
// A 16x4 f32 layout: lanes 0-15 -> M=lane, VGPR{0,1}=K{0,1}; lanes 16-31 -> K{2,3}.
// B 4x16 / C,D 16x16: row striped across lanes (N = lane&15, halves split rows).
__global__ void sage_wmma_kernel(const float* __restrict__ feat,
                                 const float* __restrict__ msg,
                                 const float* __restrict__ deg,
                                 const float* __restrict__ Wself,
                                 const float* __restrict__ Wneigh,
                                 const float* __restrict__ bself,
                                 const float* __restrict__ bneigh,
                                 const float* __restrict__ wcell,
                                 const float* __restrict__ wgene,
                                 const unsigned char* __restrict__ ntype,
                                 float* __restrict__ out) {
    const int lane = threadIdx.x & 31;
    const int wave = threadIdx.x >> 5;
    const int tileM = blockIdx.x * 4 + wave;      // 16-node tile index
    const int node0 = tileM * 16;
    if (node0 >= NN) return;                      // wave-uniform: EXEC stays all-1s
    const int ncol0 = blockIdx.y * 16;            // 16-output-dim tile index

    const int half = lane >> 4;                   // 0: K 0-1 / rows 0-7, 1: K 2-3 / rows 8-15
    const int mn   = lane & 15;                   // A: M row, B/C/D: N column
    const int kb   = half * 2;                    // K sub-offset within 4-wide step

    const float* arow_f = feat   + (size_t)(node0 + mn) * DIM;  // self A rows
    const float* arow_m = msg    + (size_t)(node0 + mn) * DIM;  // neigh A rows
    const float* brow_s = Wself  + (size_t)(ncol0 + mn) * DIM;  // B[k][n] = W[n][k]
    const float* brow_n = Wneigh + (size_t)(ncol0 + mn) * DIM;

    v8f acc_s = {};
    v8f acc_n = {};
#pragma unroll
    for (int k = 0; k < DIM; k += 4) {
        v2f a_s = *(const v2f*)(arow_f + k + kb);
        v2f b_s = *(const v2f*)(brow_s + k + kb);
        v2f a_n = *(const v2f*)(arow_m + k + kb);
        v2f b_n = *(const v2f*)(brow_n + k + kb);
        // D = A x B + C : full f32 precision matrix path on CDNA5
        acc_s = __builtin_amdgcn_wmma_f32_16x16x4_f32(
            false, a_s, false, b_s, (short)0, acc_s, false, false);
        acc_n = __builtin_amdgcn_wmma_f32_16x16x4_f32(
            false, a_n, false, b_n, (short)0, acc_n, false, false);
    }

    const float bs = bself[ncol0 + mn];
    const float bn = bneigh[ncol0 + mn];
    const float wc = wcell[0];
    const float wg = wgene[0];
#pragma unroll
    for (int i = 0; i < 8; ++i) {
        int m = node0 + i + 8 * half;             // C/D: VGPR i holds rows i and i+8
        float dv  = deg[m];
        float inv = 1.0f / fmaxf(dv, 1.0f);       // (msg/deg)@W^T == (msg@W^T)/deg
        float sw  = ntype[m] ? wc : wg;
        out[(size_t)m * DIM + ncol0 + mn] = sw * (acc_s[i] + bs) + inv * acc_n[i] + bn;
    }
}

extern "C" void kernel_launch(void* const* d_in, const int* in_sizes, int n_in,
                              void* d_out, int out_size, void* d_ws, size_t ws_size,
                              hipStream_t stream) {
    const float*         feat    = (const float*)d_in[0];
    const float*         W_self  = (const float*)d_in[1];
    const float*         b_self  = (const float*)d_in[2];
    const float*         W_neigh = (const float*)d_in[3];
    const float*         b_neigh = (const float*)d_in[4];
    const float*         w_cell  = (const float*)d_in[5];
    const float*         w_gene  = (const float*)d_in[6];
    const int*           src     = (const int*)d_in[7];
    const int*           dst     = (const int*)d_in[8];
    const unsigned char* ntype   = (const unsigned char*)d_in[9];
    float* out = (float*)d_out;

    // Workspace layout: msg_sum [NN*DIM f32] | deg [NN f32]  (~25.8 MB)
    float* msg = (float*)d_ws;
    float* deg = msg + (size_t)NN * DIM;

    // A: zero scratch (NN*DIM + NN floats; both multiples of 4)
    int n4 = (NN * DIM + NN) / 4;
    sage_zero_kernel<<<(n4 + 255) / 256, 256, 0, stream>>>((float4*)d_ws, n4);

    // B: scatter-add over edges, one wave per edge (8 edges per 256-thread block)
    int nthreads = NE * 32;
    sage_scatter_kernel<<<(nthreads + 255) / 256, 256, 0, stream>>>(
        feat, src, dst, msg, deg);

    // C: dual WMMA GEMM + combine. 3125 M-tiles (exact), 8 N-tiles, 4 waves/block.
    dim3 grid((3125 + 3) / 4, 8);
    sage_wmma_kernel<<<grid, 128, 0, stream>>>(
        feat, msg, deg, W_self, W_neigh, b_self, b_neigh, w_cell, w_gene, ntype, out);
}